// FCNRZ_5952824673078
// MI455X (gfx1250) — compile-verified
//
#include <hip/hip_runtime.h>
#include <hip/hip_bf16.h>

typedef __attribute__((ext_vector_type(2))) float v2f;
typedef __attribute__((ext_vector_type(4))) float v4f;
typedef __attribute__((ext_vector_type(8))) float v8f;

#define HISTLEN 50
#define EMB     512
#define NCHUNK  16          // EMB / 32
#define HMOD    1999968LL   // H - UMA
#define HPRIME  2147483647LL

// ---------------- weight materialization (gather) ----------------
__global__ __launch_bounds__(256)
void gather_kernel(const float* __restrict__ hw, const int* __restrict__ idx,
                   float* __restrict__ W, int n) {
  int i = blockIdx.x * 256 + threadIdx.x;
  if (i < n) W[i] = hw[idx[i]];
}

__global__ __launch_bounds__(256)
void gather_pad_kernel(const float* __restrict__ hw, const int* __restrict__ idx,
                       float* __restrict__ W, int rows, int cols, int ldw) {
  int i = blockIdx.x * 256 + threadIdx.x;
  if (i < rows * ldw) {
    int r = i / ldw, c = i - r * ldw;
    W[i] = (c < cols) ? hw[idx[r * cols + c]] : 0.0f;
  }
}

// ---------------- hashed embedding + weighted-sum pool ----------------
__global__ __launch_bounds__(512)
void embed_pool_kernel(const int* __restrict__ x, const float* __restrict__ w,
                       const float* __restrict__ hw, float* __restrict__ h0) {
  __shared__ int   s_start[HISTLEN * NCHUNK];
  __shared__ float s_w[HISTLEN];
  const int b   = blockIdx.x;
  const int tid = threadIdx.x;
  if (tid < HISTLEN) s_w[tid] = w[b * HISTLEN + tid];
  for (int i = tid; i < HISTLEN * NCHUNK; i += 512) {
    int l = i >> 4, c = i & 15;
    long long xv = (long long)x[b * HISTLEN + l];
    if (xv == 0) {
      s_start[i] = -1;  // padding_idx -> zero embedding
    } else {
      long long t = (1000003LL * xv + 193939LL * (long long)c + 7919LL) % HPRIME;
      s_start[i] = (int)(t % HMOD);
    }
  }
  __syncthreads();
  const int c = tid >> 5, o = tid & 31;  // chunk id, lane offset within chunk
  float acc = 0.0f;
#pragma unroll 10
  for (int l = 0; l < HISTLEN; ++l) {
    int st = s_start[l * NCHUNK + c];
    if (st >= 0) acc = fmaf(hw[st + o], s_w[l], acc);
  }
  h0[b * EMB + tid] = acc;
}

// ---------------- fp32 GEMM via V_WMMA_F32_16X16X4_F32 ----------------
// Workgroup tile 128x128, 8 waves (4 along M x 2 along N), each wave owns a
// 32x64 patch = 2x4 WMMA accumulators. Register-staged software pipeline:
// prefetch tile t+1 into VGPRs while tile t computes from LDS.
#define TM 128
#define TN 128
#define TK 32
#define LDA 36   // As row stride (floats): 144B, 16B-aligned for b128 fills
#define LDB 34   // Bs row stride (floats): 136B, 8B-aligned, conflict-free scatter

__global__ __launch_bounds__(256)
void gemm_f32_wmma_kernel(const float* __restrict__ A, const float* __restrict__ B,
                          float* __restrict__ C, int K, int ldb, int Nout, int ldc) {
  __shared__ float As[TM][LDA];   // 18 KB
  __shared__ float Bs[TN][LDB];   // 17 KB (transposed: Bs[n][k])
  const int tid  = threadIdx.x;
  const int lane = tid & 31;
  const int wv   = tid >> 5;      // wave 0..7
  const int wm   = wv & 3;        // 4 waves along M (32 rows each)
  const int wn   = wv >> 2;       // 2 waves along N (64 cols each)
  const int m0   = blockIdx.y * TM;
  const int n0   = blockIdx.x * TN;
  const int lr   = lane & 15;     // row/col within 16x16 tile
  const int lh   = lane >> 4;     // half-wave select
  const int lk   = lh * 2;        // base K of this half-wave's fragment pair

  // staging assignments (256 threads, float4 granularity)
  const int ar  = tid >> 3;         // A row 0..31 (+32*i), col4 = (tid&7)*4
  const int ac4 = (tid & 7) << 2;
  const int bk  = tid >> 5;         // B k-row 0..7 (+8*p), col4 = (tid&31)*4
  const int bc4 = (tid & 31) << 2;

  const size_t aRow = (size_t)32 * K;
  const size_t bRow = (size_t)8 * ldb;
  const float* aptr = A + (size_t)(m0 + ar) * K + ac4;
  const float* bptr = B + (size_t)bk * ldb + n0 + bc4;

  v8f acc[2][4] = {{{}, {}, {}, {}}, {{}, {}, {}, {}}};
  v4f areg[4], breg[4];

  // prologue: prefetch tile 0
#pragma unroll
  for (int i = 0; i < 4; ++i) areg[i] = *(const v4f*)(aptr + aRow * i);
#pragma unroll
  for (int p = 0; p < 4; ++p) breg[p] = *(const v4f*)(bptr + bRow * p);
  aptr += TK;
  bptr += (size_t)TK * ldb;

  const int ntiles = K / TK;
  for (int t = 0; t < ntiles; ++t) {
    // stage registers -> LDS
#pragma unroll
    for (int i = 0; i < 4; ++i) *(v4f*)&As[ar + 32 * i][ac4] = areg[i];
#pragma unroll
    for (int p = 0; p < 4; ++p)
#pragma unroll
      for (int j = 0; j < 4; ++j) Bs[bc4 + j][bk + 8 * p] = breg[p][j];
    __syncthreads();

    // prefetch next tile (overlaps with WMMA below)
    if (t + 1 < ntiles) {
#pragma unroll
      for (int i = 0; i < 4; ++i) areg[i] = *(const v4f*)(aptr + aRow * i);
#pragma unroll
      for (int p = 0; p < 4; ++p) breg[p] = *(const v4f*)(bptr + bRow * p);
      aptr += TK;
      bptr += (size_t)TK * ldb;
    }

    // 8 k-steps x 8 WMMA from LDS
#pragma unroll
    for (int kk = 0; kk < TK; kk += 4) {
      v2f a[2], b[4];
      a[0] = *(const v2f*)&As[wm * 32 +      lr][kk + lk];
      a[1] = *(const v2f*)&As[wm * 32 + 16 + lr][kk + lk];
#pragma unroll
      for (int nt = 0; nt < 4; ++nt)
        b[nt] = *(const v2f*)&Bs[wn * 64 + nt * 16 + lr][kk + lk];
#pragma unroll
      for (int mt = 0; mt < 2; ++mt)
#pragma unroll
        for (int nt = 0; nt < 4; ++nt)
          acc[mt][nt] = __builtin_amdgcn_wmma_f32_16x16x4_f32(
              false, a[mt], false, b[nt], (short)0, acc[mt][nt], false, false);
    }
    __syncthreads();
  }

  // write back: C element (M = v + 8*lh, N = lr) lives in acc VGPR v
#pragma unroll
  for (int mt = 0; mt < 2; ++mt) {
    const int rbase = m0 + wm * 32 + mt * 16 + 8 * lh;
#pragma unroll
    for (int nt = 0; nt < 4; ++nt) {
      const int cb = n0 + wn * 64 + nt * 16 + lr;
      if (cb < Nout) {
#pragma unroll
        for (int v = 0; v < 8; ++v)
          C[(size_t)(rbase + v) * ldc + cb] = acc[mt][nt][v];
      }
    }
  }
}

// ---------------- launcher ----------------
extern "C" void kernel_launch(void* const* d_in, const int* in_sizes, int n_in,
                              void* d_out, int out_size, void* d_ws, size_t ws_size,
                              hipStream_t stream) {
  const int*   x    = (const int*)d_in[0];
  const float* w    = (const float*)d_in[1];
  const float* hw   = (const float*)d_in[2];
  const int*   idx1 = (const int*)d_in[3];
  const int*   idx2 = (const int*)d_in[4];
  const int*   idx3 = (const int*)d_in[5];
  float* out = (float*)d_out;

  // workspace layout (64 MB total, all fp32)
  float* W1 = (float*)d_ws;            // 512  x 2048
  float* W2 = W1 + 512 * 2048;         // 2048 x 2048
  float* W3 = W2 + 2048 * 2048;        // 2048 x 1024 (1000 cols + zero pad)
  float* h0 = W3 + 2048 * 1024;        // 2048 x 512
  float* h1 = h0 + 2048 * 512;         // 2048 x 2048
  float* h2 = h1 + 2048 * 2048;        // 2048 x 2048

  gather_kernel<<<(512 * 2048 + 255) / 256, 256, 0, stream>>>(hw, idx1, W1, 512 * 2048);
  gather_kernel<<<(2048 * 2048 + 255) / 256, 256, 0, stream>>>(hw, idx2, W2, 2048 * 2048);
  gather_pad_kernel<<<(2048 * 1024 + 255) / 256, 256, 0, stream>>>(hw, idx3, W3, 2048, 1000, 1024);

  embed_pool_kernel<<<2048, 512, 0, stream>>>(x, w, hw, h0);

  dim3 blk(256);
  dim3 g12(2048 / TN, 2048 / TM);
  // h1 = h0 @ W1   (K=512)
  gemm_f32_wmma_kernel<<<g12, blk, 0, stream>>>(h0, W1, h1, 512, 2048, 2048, 2048);
  // h2 = h1 @ W2   (K=2048)
  gemm_f32_wmma_kernel<<<g12, blk, 0, stream>>>(h1, W2, h2, 2048, 2048, 2048, 2048);
  // out = h2 @ W3  (K=2048, padded ldb=1024, real N=1000)
  dim3 g3(1024 / TN, 2048 / TM);
  gemm_f32_wmma_kernel<<<g3, blk, 0, stream>>>(h2, W3, out, 2048, 1024, 1000, 1000);
}